// GiddLoss_13486197309642
// MI455X (gfx1250) — compile-verified
//
#include <hip/hip_runtime.h>
#include <hip/hip_bf16.h>
#include <stdint.h>

// ---------------- constants ----------------
#define VOCAB    50257
#define MASK_ID  50256
#define NV       50256        // non-mask vocab entries: indices 0..50255 (MASK is last)
#define BLOCK    1024         // 32 wave32 waves per workgroup
#define LOG2E    1.4426950408889634f
#define LN2      0.6931471805599453f
#define LOG_CLIP -1000000.0f
#define NEG_HUGE -3.402823466e38f

// chunk pipeline: 7 chunks of 2048 float4 (8192 dwords = 32KB) cover up to 50260 dwords
#define CH4      2048
#define CHD      8192         // dwords per chunk (= CH4*4), 2^13
#define NCH      7
#define SCRATCH_OFF 50260     // float offset of reduction scratch in LDS

// ---------------- fast transcendentals (v_exp_f32 / v_log_f32) ----------------
#if __has_builtin(__builtin_amdgcn_exp2f)
__device__ __forceinline__ float fast_exp2(float x) { return __builtin_amdgcn_exp2f(x); }
#else
__device__ __forceinline__ float fast_exp2(float x) { return exp2f(x); }
#endif
#if __has_builtin(__builtin_amdgcn_logf)
__device__ __forceinline__ float fast_log2(float x) { return __builtin_amdgcn_logf(x); }
#else
__device__ __forceinline__ float fast_log2(float x) { return log2f(x); }
#endif
__device__ __forceinline__ float fast_expn(float x) { return fast_exp2(x * LOG2E); }
__device__ __forceinline__ float fast_logn(float x) { return LN2 * fast_log2(x); }

// ---------------- CDNA5 async global->LDS path ----------------
typedef int v4i __attribute__((ext_vector_type(4)));
typedef __attribute__((address_space(1))) int  gint;   // global int
typedef __attribute__((address_space(3))) int  lint;   // LDS int
typedef __attribute__((address_space(1))) v4i  gv4i;   // global int4
typedef __attribute__((address_space(3))) v4i  lv4i;   // LDS int4

#if __has_builtin(__builtin_amdgcn_global_load_async_to_lds_b32)
#define HAVE_ASYNC_LDS 1
#define ASYNC_B32(gp, lp) __builtin_amdgcn_global_load_async_to_lds_b32((gp), (lp), 0, 0)
#if __has_builtin(__builtin_amdgcn_global_load_async_to_lds_b128)
#define HAVE_B128 1
#define ASYNC_B128(gp, lp) __builtin_amdgcn_global_load_async_to_lds_b128((gp), (lp), 0, 0)
#endif
#endif

#if defined(HAVE_B128)
#define IN_FLIGHT 2           // async instrs per wave per interior chunk
#elif defined(HAVE_ASYNC_LDS)
#define IN_FLIGHT 8
#endif

#if defined(HAVE_ASYNC_LDS)
#if __has_builtin(__builtin_amdgcn_s_wait_asynccnt)
#define WAITN(n) __builtin_amdgcn_s_wait_asynccnt(n)
#else
#define WAITN(n) asm volatile("s_wait_asynccnt %0" ::"i"(n) : "memory")
#endif
#else
#define WAITN(n)
#endif

// ---------------- deterministic reductions (wave32 + cross-wave LDS) ----------------
__device__ __forceinline__ float wred_sum(float v) {
#pragma unroll
  for (int m = 16; m >= 1; m >>= 1) v += __shfl_xor(v, m, 32);
  return v;
}
__device__ __forceinline__ float wred_max(float v) {
#pragma unroll
  for (int m = 16; m >= 1; m >>= 1) v = fmaxf(v, __shfl_xor(v, m, 32));
  return v;
}
// block-wide reduce + broadcast; `red` needs >= 32 floats; nwaves = blockDim/32
__device__ __forceinline__ float bred(float v, float* red, int nwaves, bool do_max) {
  v = do_max ? wred_max(v) : wred_sum(v);
  const int wave = threadIdx.x >> 5;
  const int lane = threadIdx.x & 31;
  __syncthreads();                       // protect `red` from previous use
  if (lane == 0) red[wave] = v;
  __syncthreads();
  if (wave == 0) {
    float x = (lane < nwaves) ? red[lane] : (do_max ? NEG_HUGE : 0.0f);
    x = do_max ? wred_max(x) : wred_sum(x);
    if (lane == 0) red[0] = x;
  }
  __syncthreads();
  return red[0];
}

// ---------------- kernel 1: one token per workgroup ----------------
__global__ __launch_bounds__(BLOCK)
void gidd_token_kernel(const float* __restrict__ logits,
                       const int*   __restrict__ input_ids,
                       const float* __restrict__ amask,
                       const int*   __restrict__ z_t,
                       const float* __restrict__ t_arr,
                       const float* __restrict__ log_gamma_p,
                       const float* __restrict__ log_B_p,
                       int S,
                       float* __restrict__ elbo_out,
                       float* __restrict__ partials) {
  extern __shared__ float lds[];            // SCRATCH_OFF data floats + 48 scratch
  float* red = lds + SCRATCH_OFF;
  const int token = blockIdx.x;
  const int tid   = threadIdx.x;

  // staging window: align global start down to 16B so b128 works on both sides
  const int pad4  = token & 3;              // head junk dwords
  const int D     = pad4 + NV;              // dwords from aligned start to row end
  const int full4 = D >> 2;                 // full float4s of loaded data
  const int rpad  = (4 - (D & 3)) & 3;      // tail junk dwords (never loaded)
  const int NT4   = (D + 3) >> 2;           // float4s swept by compute

#if defined(HAVE_ASYNC_LDS)
  gint* g0 = (gint*)(uintptr_t)(logits + (size_t)token * VOCAB - pad4);
  lint* l0 = (lint*)lds;
#else
  const float* g0f = logits + (size_t)token * VOCAB - pad4;
  for (int d = tid; d < D; d += BLOCK) lds[d] = g0f[d];
#endif
  float4* L4 = (float4*)lds;

#if defined(HAVE_ASYNC_LDS)
  auto issue = [&](int c) {
#if defined(HAVE_B128)
    gv4i* g4 = (gv4i*)g0;
    lv4i* l4 = (lv4i*)l0;
    int e4 = (c + 1) * CH4 < full4 ? (c + 1) * CH4 : full4;
    for (int q = c * CH4 + tid; q < e4; q += BLOCK)
      ASYNC_B128(g4 + q, l4 + q);
#else
    int ed = (c + 1) * CHD < full4 * 4 ? (c + 1) * CHD : full4 * 4;
    for (int d = c * CHD + tid; d < ed; d += BLOCK)
      ASYNC_B32(g0 + d, l0 + d);
#endif
    if (c == NCH - 1) {                     // leftover row dwords [4*full4, D)
      int d = 4 * full4 + tid;
      if (d < D) ASYNC_B32(g0 + d, l0 + d);
    }
  };
#else
  auto issue = [&](int) {};
#endif

  // ---- per-token mixture constants (compute while DMA flies) ----
  const float tb    = t_arr[token / S];
  const float gamma = fast_expn(log_gamma_p[0]);
  const float Bc    = fast_expn(log_B_p[0]);
  const float omt   = 1.0f - tb;
  const float c_t   = fast_exp2(0.5f * gamma * fast_log2(tb * omt)) * Bc;
  const float Dn    = 1.0f + c_t;
  const float a     = omt / Dn;                    // weight of x_hat component
  const float bq    = c_t / ((float)NV * Dn);      // uniform component per entry

  // ---- pipelined fill + per-chunk (max, exp-store, sum) ----
  float mch[NCH], zch[NCH];
  issue(0);
#pragma unroll
  for (int c = 0; c < NCH; ++c) {
    if (c + 1 < NCH) issue(c + 1);
    if (c + 2 < NCH) { WAITN(IN_FLIGHT); } else { WAITN(0); }
    __syncthreads();                       // chunk c resident for all waves
    if (c == 0 || c == NCH - 1) {          // neutralize junk dwords
      if (c == 0 && tid < pad4) lds[tid] = NEG_HUGE;
      if (c == NCH - 1 && tid < rpad) lds[D + tid] = NEG_HUGE;
      __syncthreads();
    }
    const int s4 = c * CH4;
    const int e4 = (c + 1) * CH4 < NT4 ? (c + 1) * CH4 : NT4;
    float mloc = NEG_HUGE;
    for (int q = s4 + tid; q < e4; q += BLOCK) {
      float4 x = L4[q];
      mloc = fmaxf(fmaxf(mloc, fmaxf(x.x, x.y)), fmaxf(x.z, x.w));
    }
    const float mc = bred(mloc, red, BLOCK / 32, true);
    float zloc = 0.0f;
    for (int q = s4 + tid; q < e4; q += BLOCK) {
      float4 x = L4[q];
      float4 e;
      e.x = fast_exp2((x.x - mc) * LOG2E);
      e.y = fast_exp2((x.y - mc) * LOG2E);
      e.z = fast_exp2((x.z - mc) * LOG2E);
      e.w = fast_exp2((x.w - mc) * LOG2E);
      L4[q] = e;
      zloc += (e.x + e.y) + (e.z + e.w);
    }
    zch[c] = bred(zloc, red, BLOCK / 32, false);
    mch[c] = mc;
  }

  // ---- combine chunk stats: global max, Z, per-chunk scales ----
  float M = mch[0];
#pragma unroll
  for (int c = 1; c < NCH; ++c) M = fmaxf(M, mch[c]);
  float Z = 0.0f;
  float as[NCH];
#pragma unroll
  for (int c = 0; c < NCH; ++c) {
    const float sc = fast_exp2((mch[c] - M) * LOG2E);
    Z += zch[c] * sc;
    as[c] = a * sc;
  }
  const float bqZ = bq * Z;
  const float lnZ = LN2 * fast_log2(Z);
  if (tid == 0) {                          // publish scales for epilogue gathers
#pragma unroll
    for (int c = 0; c < NCH; ++c) red[32 + c] = as[c];
  }

  // ---- pass C: S1 = sum_v clip(log(a*xhat_v + bq)) ----
  float s1 = 0.0f;
#pragma unroll
  for (int c = 0; c < NCH; ++c) {
    const float ac = as[c];
    const int s4 = c * CH4;
    const int e4 = (c + 1) * CH4 < NT4 ? (c + 1) * CH4 : NT4;
    for (int q = s4 + tid; q < e4; q += BLOCK) {
      float4 x = L4[q];
      float v0 = fmaxf(fmaf(LN2, fast_log2(fmaf(ac, x.x, bqZ)), -lnZ), LOG_CLIP);
      float v1 = fmaxf(fmaf(LN2, fast_log2(fmaf(ac, x.y, bqZ)), -lnZ), LOG_CLIP);
      float v2 = fmaxf(fmaf(LN2, fast_log2(fmaf(ac, x.z, bqZ)), -lnZ), LOG_CLIP);
      float v3 = fmaxf(fmaf(LN2, fast_log2(fmaf(ac, x.w, bqZ)), -lnZ), LOG_CLIP);
      s1 += (v0 + v1) + (v2 + v3);
    }
  }
  float S1 = bred(s1, red, BLOCK / 32, false);
  // remove junk contributions: e=0 slots computed exactly fmaf(LN2,log2(bqZ),-lnZ)
  const float val0 = fmaxf(fmaf(LN2, fast_log2(bqZ), -lnZ), LOG_CLIP);
  S1 -= (float)(pad4 + rpad) * val0;

  // ---- per-token epilogue ----
  if (tid == 0) {
    const int id = input_ids[token];
    const int z  = z_t[token];

    const float lq_else = fmaxf(fast_logn(bq), LOG_CLIP);
    const float lq_id   = fmaxf(fast_logn(a + bq), LOG_CLIP);
    const float e_id  = lds[pad4 + id];
    const float as_id = red[32 + ((pad4 + id) >> 13)];
    const float L_id  = fmaxf(fmaf(LN2, fast_log2(fmaf(as_id, e_id, bqZ)), -lnZ), LOG_CLIP);

    // KL: uniform bulk + corrected id term; MASK term cancels exactly (q==p there)
    const float kl = bq * ((float)(NV - 1) * lq_else - (S1 - L_id))
                   + (a + bq) * (lq_id - L_id);

    // importance-sampling term at z_t (zero when z_t == MASK: q==p)
    float isl = 0.0f;
    if (z != MASK_ID) {
      const float e_z  = lds[pad4 + z];
      const float as_z = red[32 + ((pad4 + z) >> 13)];
      const float lp_z = fmaxf(fmaf(LN2, fast_log2(fmaf(as_z, e_z, bqZ)), -lnZ), LOG_CLIP);
      const float lq_z = (z == id) ? lq_id : lq_else;
      const float r    = lq_z - lp_z;
      isl = fast_expn(r) - r - 1.0f;
    }

    // weights
    const float ctp  = 0.5f * gamma * (1.0f - 2.0f * tb) / (tb * omt) * c_t;
    const float num  = c_t + omt * ctp;
    const float w_x  = num / (float)NV / (omt * (omt + c_t / (float)NV));
    const float w_u  = num / (omt * c_t);
    const float w_m  = 1.0f / (omt * tb);
    const bool is_mask = (z == MASK_ID);
    const bool is_x    = (z == id);
    const float elbo_w = is_x ? w_x : (is_mask ? w_m : w_u);
    const float sig     = 1.0f / (1.0f + fast_expn(tb));        // sigmoid(-t)
    const float x_scale = Bc / (float)VOCAB * fast_expn(0.5f * gamma * sig);
    float loss_w = is_x ? x_scale : (is_mask ? 2.0f : 1.0f);
    const float ws = fminf(fmaxf(loss_w, 0.0f), 2.0f);

    const float tot  = kl + isl;
    const float elbo = elbo_w * tot;
    elbo_out[token]  = elbo;

    const float msk = amask[token];
    float* p = partials + 6 * token;
    p[0] = ws * tot;        // -> loss numerator
    p[1] = ws * msk;        // -> denom_ws
    p[2] = ws * kl  * msk;  // -> kl_m numerator
    p[3] = ws * isl * msk;  // -> is_m numerator
    p[4] = elbo * msk;      // -> elbo_m numerator
    p[5] = msk;             // -> mask sum
  }
}

// ---------------- kernel 2: deterministic final reduction ----------------
__global__ __launch_bounds__(BLOCK)
void gidd_finalize(const float* __restrict__ partials, int tokens,
                   float* __restrict__ out) {
  __shared__ float red[32];
  float s[6] = {0.f, 0.f, 0.f, 0.f, 0.f, 0.f};
  for (int tk = threadIdx.x; tk < tokens; tk += BLOCK) {
    const float* p = partials + 6 * tk;
#pragma unroll
    for (int i = 0; i < 6; i++) s[i] += p[i];
  }
  float tot[6];
#pragma unroll
  for (int i = 0; i < 6; i++) tot[i] = bred(s[i], red, BLOCK / 32, false);
  if (threadIdx.x == 0) {
    const float eps = 1.1920929e-07f;              // f32 machine eps
    out[0] = tot[0] / (float)tokens;               // loss
    const float denom = fmaxf(tot[1], eps);
    out[1 + tokens + 0] = tot[2] / denom;          // kl_m
    out[1 + tokens + 1] = tot[3] / denom;          // is_m
    out[1 + tokens + 2] = tot[4] / fmaxf(tot[5], eps); // elbo_m
  }
}

// ---------------- host launch ----------------
extern "C" void kernel_launch(void* const* d_in, const int* in_sizes, int n_in,
                              void* d_out, int out_size, void* d_ws, size_t ws_size,
                              hipStream_t stream) {
  const float* logits    = (const float*)d_in[0];
  const int*   input_ids = (const int*)d_in[1];
  const float* amask     = (const float*)d_in[2];
  const int*   z_t       = (const int*)d_in[3];
  const float* t         = (const float*)d_in[4];
  const float* lg        = (const float*)d_in[5];
  const float* lB        = (const float*)d_in[6];
  float* out = (float*)d_out;

  const int tokens = in_sizes[1];        // B*S
  const int B      = in_sizes[4];        // t has shape (B,)
  const int S      = tokens / B;

  float* partials = (float*)d_ws;        // 6 floats per token

  const size_t smem = (size_t)(SCRATCH_OFF + 48) * sizeof(float);
  gidd_token_kernel<<<tokens, BLOCK, smem, stream>>>(
      logits, input_ids, amask, z_t, t, lg, lB, S, out + 1, partials);
  gidd_finalize<<<1, BLOCK, 0, stream>>>(partials, tokens, out);
}